// AttentionAdapter_79319456022725
// MI455X (gfx1250) — compile-verified
//
#include <hip/hip_runtime.h>
#include <math.h>

// 128-bit vector type for B128 global loads/stores
typedef float __attribute__((ext_vector_type(4))) f4;

#define COPY_THREADS 256
#define VECS_PER_THREAD 4
#define VECS_PER_BLOCK (COPY_THREADS * VECS_PER_THREAD)   // 1024 float4 = 16 KB

// ---------------------------------------------------------------------------
// Hot kernel: branch-free streaming NT copy. Host guarantees every block is
// full, so there are no bounds checks -> no EXEC-mask branches, no zero-fill
// v_movs. Body = address math + 4x global_load_b128 th:TH_LOAD_NT +
// 4x global_store_b128 th:TH_STORE_NT. The 1 GB stream >> 192 MB L2, so NT
// keeps L2 clean; ILP=4 keeps LOADcnt/STOREcnt deep per lane.
// ---------------------------------------------------------------------------
__global__ void __launch_bounds__(COPY_THREADS)
nt_copy_b128_kernel(const f4* __restrict__ src, f4* __restrict__ dst) {
    const long long base =
        (long long)blockIdx.x * VECS_PER_BLOCK + threadIdx.x;

    f4 v0 = __builtin_nontemporal_load(&src[base + 0 * COPY_THREADS]);
    f4 v1 = __builtin_nontemporal_load(&src[base + 1 * COPY_THREADS]);
    f4 v2 = __builtin_nontemporal_load(&src[base + 2 * COPY_THREADS]);
    f4 v3 = __builtin_nontemporal_load(&src[base + 3 * COPY_THREADS]);

    __builtin_nontemporal_store(v0, &dst[base + 0 * COPY_THREADS]);
    __builtin_nontemporal_store(v1, &dst[base + 1 * COPY_THREADS]);
    __builtin_nontemporal_store(v2, &dst[base + 2 * COPY_THREADS]);
    __builtin_nontemporal_store(v3, &dst[base + 3 * COPY_THREADS]);
}

// ---------------------------------------------------------------------------
// Tail kernel: scalar copy of whatever the full blocks didn't cover.
// (For the reference shape, total = 2^27 floats, remainder = 0 and this
// never launches; kept for generality/safety.)
// ---------------------------------------------------------------------------
__global__ void tail_copy_kernel(const float* __restrict__ src,
                                 float* __restrict__ dst,
                                 long long start, long long total) {
    long long i = start + (long long)blockIdx.x * blockDim.x + threadIdx.x;
    if (i < total) dst[i] = src[i];
}

// ---------------------------------------------------------------------------
// Fixup kernel: sparse scatter-multiply, 512 lanes total.
// out[h, final_poss[d], class_poss[d]] *= exp(weight[h, d])
// Stream-ordered after the copy, so it reads freshly written values.
// ---------------------------------------------------------------------------
__global__ void scatter_mul_kernel(float* __restrict__ out,
                                   const float* __restrict__ weight,
                                   const long long* __restrict__ class_poss,
                                   const long long* __restrict__ final_poss,
                                   int n_head, int n_demo, long long seq) {
    int t = blockIdx.x * blockDim.x + threadIdx.x;
    if (t >= n_head * n_demo) return;
    int h = t / n_demo;
    int d = t % n_demo;
    long long row = final_poss[d];
    long long col = class_poss[d];
    long long idx = ((long long)h * seq + row) * seq + col;
    out[idx] *= expf(weight[h * n_demo + d]);
}

extern "C" void kernel_launch(void* const* d_in, const int* in_sizes, int n_in,
                              void* d_out, int out_size, void* d_ws, size_t ws_size,
                              hipStream_t stream) {
    const float*     attn   = (const float*)d_in[0];      // (1, n_head, seq, seq) f32
    const float*     weight = (const float*)d_in[1];      // (n_head, n_demo)      f32
    const long long* cls    = (const long long*)d_in[2];  // (n_demo,)             i64
    const long long* fin    = (const long long*)d_in[3];  // (n_demo,)             i64
    float*           out    = (float*)d_out;

    const long long total  = (long long)in_sizes[0];
    const int       n_demo = in_sizes[2];
    const int       n_head = in_sizes[1] / n_demo;
    const long long seq2   = total / n_head;               // batch == 1
    const long long seq    = (long long)(sqrt((double)seq2) + 0.5);

    // ---- streaming NT copy: full blocks only, branch-free hot path ----
    const long long nvec4       = total / 4;
    const long long full_blocks = nvec4 / VECS_PER_BLOCK;
    if (full_blocks > 0) {
        nt_copy_b128_kernel<<<(int)full_blocks, COPY_THREADS, 0, stream>>>(
            (const f4*)attn, (f4*)out);
    }

    // ---- tail: everything not covered by full blocks (0 for 2^27) ----
    const long long covered = full_blocks * (long long)VECS_PER_BLOCK * 4;
    const long long tail_n  = total - covered;
    if (tail_n > 0) {
        int tb = (int)((tail_n + 255) / 256);
        tail_copy_kernel<<<tb, 256, 0, stream>>>(attn, out, covered, total);
    }

    // ---- sparse fixup: 512 elements scaled by exp(weight) ----
    const int nfix = n_head * n_demo;
    if (nfix > 0) {
        int fb = (nfix + 255) / 256;
        scatter_mul_kernel<<<fb, 256, 0, stream>>>(out, weight, cls, fin,
                                                   n_head, n_demo, seq);
    }
}